// stableMIL_86655260164951
// MI455X (gfx1250) — compile-verified
//
#include <hip/hip_runtime.h>
#include <math.h>

// Problem constants (match reference)
#define N_TOK  131072
#define C_DIM  768
#define N1_GRP 32768
#define H_DIM  512

typedef float v2f __attribute__((ext_vector_type(2)));
typedef float v8f __attribute__((ext_vector_type(8)));

// ---------------------------------------------------------------------------
// Kernel 1: starts[g] = lower_bound(labels, g); starts[N1] = N.
// Labels are sorted with every bin non-empty, so each group is a contiguous
// row range [starts[g], starts[g+1]).
// ---------------------------------------------------------------------------
__global__ void k_starts(const int* __restrict__ labels, int* __restrict__ starts) {
  int g = blockIdx.x * blockDim.x + threadIdx.x;
  if (g > N1_GRP) return;
  if (g == N1_GRP) { starts[g] = N_TOK; return; }
  int lo = 0, hi = N_TOK;
  while (lo < hi) {
    int mid = (lo + hi) >> 1;
    if (labels[mid] < g) lo = mid + 1; else hi = mid;
  }
  starts[g] = lo;
}

// ---------------------------------------------------------------------------
// Kernel 2: per-group fused feature (memory-bound phase: 402MB in, 100MB out).
// One wave32 per group; lane covers 6 float4 column slices (fully coalesced).
// feats = (sum_first_min(4,c) + (4 - min(4,c)) * mean) / 4
// ---------------------------------------------------------------------------
__global__ void k_feats(const float* __restrict__ x,
                        const int*   __restrict__ sidx,
                        const int*   __restrict__ starts,
                        float*       __restrict__ feats) {
  const int lane = threadIdx.x & 31;
  const int wv   = threadIdx.x >> 5;
  const int g    = blockIdx.x * 8 + wv;
  const int s    = starts[g];
  const int e    = starts[g + 1];
  const int cnt  = e - s;

  float4 tot[6], f4[6];
#pragma unroll
  for (int j = 0; j < 6; ++j) {
    tot[j] = make_float4(0.f, 0.f, 0.f, 0.f);
    f4[j]  = make_float4(0.f, 0.f, 0.f, 0.f);
  }

  for (int r = s; r < e; ++r) {
    const int row = sidx[r];  // identity in practice, but stay faithful
    const float4* src = (const float4*)(x + (size_t)row * C_DIM) + lane;
    const bool in4 = (r - s) < 4;
#pragma unroll
    for (int j = 0; j < 6; ++j) {
      float4 v = src[j * 32];
      tot[j].x += v.x; tot[j].y += v.y; tot[j].z += v.z; tot[j].w += v.w;
      if (in4) { f4[j].x += v.x; f4[j].y += v.y; f4[j].z += v.z; f4[j].w += v.w; }
    }
  }

  const float vcnt  = (float)(cnt < 4 ? cnt : 4);
  const float wmean = (4.0f - vcnt) * (1.0f / (float)cnt) * 0.25f;
  float4* dst = (float4*)(feats + (size_t)g * C_DIM) + lane;
#pragma unroll
  for (int j = 0; j < 6; ++j) {
    dst[j * 32] = make_float4(f4[j].x * 0.25f + tot[j].x * wmean,
                              f4[j].y * 0.25f + tot[j].y * wmean,
                              f4[j].z * 0.25f + tot[j].z * wmean,
                              f4[j].w * 0.25f + tot[j].w * wmean);
  }
}

// ---------------------------------------------------------------------------
// Kernel 3: out = gelu(feats @ W + b), fp32 WMMA (V_WMMA_F32_16X16X4_F32).
// Block = 256 threads (8 waves) -> 128x64 output tile; wave w owns rows
// [w*16, w*16+16) x 64 cols = four 16x16 C tiles. K staged 32-wide via LDS.
// A fragment layout (16x4 f32): lane = M, K = 2*(lane>>4) + vgpr -> contiguous
// float2 from row-major LDS A (pad stride 36 dwords, conflict-free).
// B fragment: lane = N, same K mapping -> store B k-pair-interleaved in LDS so
// (k, k+1) is one contiguous float2.
// ---------------------------------------------------------------------------
#define BM  128
#define BN  64
#define KC  32
#define LDA 36

__launch_bounds__(256)
__global__ void k_gemm_gelu(const float* __restrict__ A,     // feats [N1, C]
                            const float* __restrict__ W,     // [C, H]
                            const float* __restrict__ bias,  // [H]
                            float*       __restrict__ out) { // [N1, H]
  __shared__ float Asm[BM * LDA];            // 128 x 32 (+pad)   = 18.0 KB
  __shared__ float Bsm[(KC / 2) * (BN * 2)]; // 16 pairs x 128    =  8.0 KB

  const int tid  = threadIdx.x;
  const int lane = tid & 31;
  const int wv   = tid >> 5;
  const int hi   = lane >> 4;   // lane half: selects K bit1 (A/B), M bit3 (C)
  const int lm   = lane & 15;   // M for A-frag, N for B/C-frag
  const int rb   = blockIdx.y * BM;
  const int cb   = blockIdx.x * BN;

  const v8f vzero = {0.f, 0.f, 0.f, 0.f, 0.f, 0.f, 0.f, 0.f};
  v8f acc[4];
#pragma unroll
  for (int t = 0; t < 4; ++t) acc[t] = vzero;

  for (int k0 = 0; k0 < C_DIM; k0 += KC) {
    __syncthreads();
    // Stage A: 128x32 fp32, coalesced float4 loads
#pragma unroll
    for (int p = 0; p < 4; ++p) {
      int lin = p * 256 + tid;
      int r   = lin >> 3;          // 0..127
      int kq  = (lin & 7) << 2;    // 0,4,...,28
      float4 v = *(const float4*)(A + (size_t)(rb + r) * C_DIM + k0 + kq);
      *(float4*)(&Asm[r * LDA + kq]) = v;
    }
    // Stage B: 32x64 fp32, write k-pair interleaved [pair][col*2 + (k&1)]
#pragma unroll
    for (int p = 0; p < 2; ++p) {
      int lin = p * 256 + tid;
      int k   = lin >> 4;          // 0..31
      int cq  = (lin & 15) << 2;   // col base 0..60
      float4 v = *(const float4*)(W + (size_t)(k0 + k) * H_DIM + cb + cq);
      float* bd = &Bsm[(k >> 1) * (BN * 2) + (k & 1)];
      bd[(cq + 0) * 2] = v.x;
      bd[(cq + 1) * 2] = v.y;
      bd[(cq + 2) * 2] = v.z;
      bd[(cq + 3) * 2] = v.w;
    }
    __syncthreads();

    const float* arow = &Asm[(wv * 16 + lm) * LDA];
#pragma unroll
    for (int s = 0; s < KC; s += 4) {
      v2f a = *(const v2f*)(arow + s + 2 * hi);               // K = s+2*hi, +1
      const float* bk = &Bsm[((s >> 1) + hi) * (BN * 2) + lm * 2];
#pragma unroll
      for (int t = 0; t < 4; ++t) {
        v2f b = *(const v2f*)(bk + t * 32);                   // cols t*16+lm
        acc[t] = __builtin_amdgcn_wmma_f32_16x16x4_f32(
            false, a, false, b, (short)0, acc[t], false, false);
      }
    }
  }

  // Epilogue: bias + exact-erf GELU, C layout: M = vgpr + 8*hi, N = lm
  const int orow0 = rb + wv * 16 + hi * 8;
#pragma unroll
  for (int t = 0; t < 4; ++t) {
    const int col = cb + t * 16 + lm;
    const float bv = bias[col];
#pragma unroll
    for (int i = 0; i < 8; ++i) {
      float xv = acc[t][i] + bv;
      float gv = 0.5f * xv * (1.0f + erff(xv * 0.70710678118654752f));
      out[(size_t)(orow0 + i) * H_DIM + col] = gv;
    }
  }
}

// ---------------------------------------------------------------------------
// Launch: workspace = [ starts: (N1+1) ints | pad | feats: N1*C floats ]
// Needs ~100.8 MB of d_ws.
// ---------------------------------------------------------------------------
extern "C" void kernel_launch(void* const* d_in, const int* in_sizes, int n_in,
                              void* d_out, int out_size, void* d_ws, size_t ws_size,
                              hipStream_t stream) {
  const float* x      = (const float*)d_in[0];
  // d_in[1] = coords (unused by reference output)
  const int*   labels = (const int*)d_in[2];
  const int*   sidx   = (const int*)d_in[3];
  const float* Wmap   = (const float*)d_in[4];
  const float* bmap   = (const float*)d_in[5];
  float*       out    = (float*)d_out;

  int*   starts = (int*)d_ws;
  float* feats  = (float*)((char*)d_ws + 131328);  // (N1+1)*4 rounded to 256B

  k_starts<<<(N1_GRP + 1 + 255) / 256, 256, 0, stream>>>(labels, starts);
  k_feats<<<N1_GRP / 8, 256, 0, stream>>>(x, sidx, starts, feats);
  k_gemm_gelu<<<dim3(H_DIM / BN, N1_GRP / BM), 256, 0, stream>>>(feats, Wmap, bmap, out);
}